// EncoderBlock_90950227460795
// MI455X (gfx1250) — compile-verified
//
#include <hip/hip_runtime.h>
#include <math.h>

typedef __attribute__((ext_vector_type(2))) float v2f;
typedef __attribute__((ext_vector_type(8))) float v8f;

#define BATCH 16
#define NPTS  32768
#define CDIM  256
#define NSMP  64
#define KNBR  8
#define R2    16.0f

// ---------------- workspace layout (bytes, 256-aligned) ----------------
#define OFF_IDX     0u               // 1024 int
#define OFF_NIDX    4096u            // 8192 int
#define OFF_SCOOR   36864u           // 3072 f
#define OFF_SIF     49152u           // 3072 f
#define OFF_DCOOR   61440u           // 3072 f
#define OFF_DIF     73728u           // 3072 f
#define OFF_NQ      86016u           // 262144 f
#define OFF_NK      1134592u         // 262144 f
#define OFF_QH      2183168u         // 262144 f
#define OFF_KH      3231744u         // 262144 f
#define OFF_LOG     4280320u         // 65536 f  (end ~4.54 MB)

// ======================= 1) Farthest point sampling =======================
// One workgroup per batch. dists live in registers (32 pts/thread, strided).
// coor re-read per step stays L2-resident (6 MB total working set).
__global__ __launch_bounds__(1024) void fps_kernel(const float* __restrict__ coor,
                                                   int* __restrict__ idx)
{
    __shared__ float rv[1024];
    __shared__ int   ri[1024];
    __shared__ int   sLast;
    const int b = blockIdx.x;
    const int t = threadIdx.x;
    const float* cb = coor + (size_t)b * NPTS * 3;

    float d[32];
#pragma unroll
    for (int i = 0; i < 32; ++i) d[i] = 1e10f;

    if (t == 0) { sLast = 0; idx[b * NSMP] = 0; }
    __syncthreads();

    for (int s = 1; s < NSMP; ++s) {
        const int last = sLast;
        const float lx = cb[last*3+0], ly = cb[last*3+1], lz = cb[last*3+2];
        float best = -1.0f; int bi = 0;
#pragma unroll
        for (int i = 0; i < 32; ++i) {
            const int p = t + (i << 10);               // ascending within thread
            const float dx = cb[p*3+0] - lx;
            const float dy = cb[p*3+1] - ly;
            const float dz = cb[p*3+2] - lz;
            const float dd = dx*dx + dy*dy + dz*dz;
            d[i] = fminf(d[i], dd);
            if (d[i] > best) { best = d[i]; bi = p; }  // strict > keeps lowest index on tie
        }
        rv[t] = best; ri[t] = bi;
        __syncthreads();
        for (int off = 512; off > 0; off >>= 1) {
            if (t < off) {
                const float v2 = rv[t + off]; const int i2 = ri[t + off];
                if (v2 > rv[t] || (v2 == rv[t] && i2 < ri[t])) { rv[t] = v2; ri[t] = i2; }
            }
            __syncthreads();
        }
        if (t == 0) { sLast = ri[0]; idx[b * NSMP + s] = ri[0]; }
        __syncthreads();
    }
}

// ======================= 2) Ball query =======================
// Thread per center; first-8-valid-in-index-order == sort-then-take-8 of the
// reference. Center itself (d2==0) guarantees >=1 valid; early exit dominates.
__global__ void ballq_kernel(const float* __restrict__ coor,
                             const int* __restrict__ idx,
                             int* __restrict__ nidx)
{
    const int b = blockIdx.x;
    const int s = threadIdx.x;
    const float* cb = coor + (size_t)b * NPTS * 3;
    const int ci = idx[b * NSMP + s];
    const float cx = cb[ci*3+0], cy = cb[ci*3+1], cz = cb[ci*3+2];
    int out[KNBR]; int cnt = 0;
    for (int j = 0; j < NPTS && cnt < KNBR; ++j) {
        const float dx = cb[j*3+0]-cx, dy = cb[j*3+1]-cy, dz = cb[j*3+2]-cz;
        if (dx*dx + dy*dy + dz*dz < R2) out[cnt++] = j;
    }
    const int f = (cnt > 0) ? out[0] : 0;
    for (int m = cnt; m < KNBR; ++m) out[m] = f;
#pragma unroll
    for (int m = 0; m < KNBR; ++m) nidx[(b * NSMP + s) * KNBR + m] = out[m];
}

// ======================= 3) Gather + max/mean + dual LayerNorm =======================
__device__ inline float blockSum256(float v, float* red)
{
    const int t = threadIdx.x;
    red[t] = v; __syncthreads();
#pragma unroll
    for (int off = 128; off > 0; off >>= 1) {
        if (t < off) red[t] += red[t + off];
        __syncthreads();
    }
    const float r = red[0];
    __syncthreads();
    return r;
}

__global__ __launch_bounds__(256) void gather_ln_kernel(
    const float* __restrict__ x, const float* __restrict__ coor,
    const float* __restrict__ iff, const int* __restrict__ idx,
    const int* __restrict__ nidx,
    const float* __restrict__ gq, const float* __restrict__ bq,
    const float* __restrict__ gk, const float* __restrict__ bk,
    float* __restrict__ out1, float* __restrict__ nqw, float* __restrict__ nkw,
    float* __restrict__ scoor, float* __restrict__ sif,
    float* __restrict__ dcoor, float* __restrict__ dif)
{
    __shared__ float red[256];
    const int bs = blockIdx.x;
    const int b  = bs >> 6;
    const int c  = threadIdx.x;
    const int si = idx[bs];
    int nb[KNBR];
#pragma unroll
    for (int m = 0; m < KNBR; ++m) nb[m] = nidx[bs * KNBR + m];

    const float* xb = x + (size_t)b * NPTS * CDIM;
    const float sx = xb[(size_t)si * CDIM + c];
    float gmax = -INFINITY;
#pragma unroll
    for (int m = 0; m < KNBR; ++m) gmax = fmaxf(gmax, xb[(size_t)nb[m] * CDIM + c]);
    out1[(size_t)bs * CDIM + c] = gmax;           // sample_x + x_2 == global_x
    const float x2 = gmax - sx;

    // LN(sample_x) -> nk
    {
        const float mean = blockSum256(sx, red) * (1.0f / CDIM);
        const float msq  = blockSum256(sx * sx, red) * (1.0f / CDIM);
        const float inv  = rsqrtf(msq - mean * mean + 1e-5f);
        nkw[(size_t)bs * CDIM + c] = (sx - mean) * inv * gk[c] + bk[c];
    }
    // LN(x_2) -> nq
    {
        const float mean = blockSum256(x2, red) * (1.0f / CDIM);
        const float msq  = blockSum256(x2 * x2, red) * (1.0f / CDIM);
        const float inv  = rsqrtf(msq - mean * mean + 1e-5f);
        nqw[(size_t)bs * CDIM + c] = (x2 - mean) * inv * gq[c] + bq[c];
    }

    if (c < 3) {
        const float* cb = coor + (size_t)b * NPTS * 3;
        const float* fb = iff  + (size_t)b * NPTS * 3;
        const float scv = cb[(size_t)si * 3 + c];
        const float sfv = fb[(size_t)si * 3 + c];
        float sc = 0.0f, sf = 0.0f;
#pragma unroll
        for (int m = 0; m < KNBR; ++m) { sc += cb[(size_t)nb[m]*3 + c]; sf += fb[(size_t)nb[m]*3 + c]; }
        scoor[bs*3 + c] = scv;
        sif  [bs*3 + c] = sfv;
        dcoor[bs*3 + c] = sc * 0.125f - scv;      // mean(coor_nbr - center)
        dif  [bs*3 + c] = sf * 0.125f;            // mean(if_nbr)
    }
}

// ======================= WMMA fp32 16x16 tile, K=256 =======================
// D[m][n] = sum_k A[m][k] * Bm[n][k]   (Bm consumed transposed)
// V_WMMA_F32_16X16X4_F32 layouts: A lanes0-15:{K0,K1}, lanes16-31:{K2,K3};
// B mirrored; C/D VGPR r: lanes0-15 M=r, lanes16-31 M=r+8.
__device__ inline void wmma_tile_f32(const float* __restrict__ A, int lda,
                                     const float* __restrict__ Bm, int ldb,
                                     float* __restrict__ D, int ldd,
                                     int mBase, int nBase, int Kdim, int lane)
{
    const int lm = lane & 15;
    const int hi = lane >> 4;           // 0 or 1
    const float* arow = A  + (size_t)(mBase + lm) * lda;
    const float* brow = Bm + (size_t)(nBase + lm) * ldb;
    v8f acc = {};
    for (int k0 = 0; k0 < Kdim; k0 += 4) {
        v2f av, bv;
        av.x = arow[k0 + 2*hi];
        av.y = arow[k0 + 2*hi + 1];
        bv.x = brow[k0 + 2*hi];
        bv.y = brow[k0 + 2*hi + 1];
        acc = __builtin_amdgcn_wmma_f32_16x16x4_f32(
            /*neg_a=*/false, av, /*neg_b=*/false, bv,
            /*c_mod=*/(short)0, acc, /*reuse_a=*/false, /*reuse_b=*/false);
    }
#pragma unroll
    for (int r = 0; r < 8; ++r)
        D[(size_t)(mBase + r + 8*hi) * ldd + nBase + lm] = acc[r];
}

// ---- 4) qh = nq @ Wq^T ; kh = nk @ Wk^T (per batch 64x256, K=256) ----
__global__ __launch_bounds__(256) void gemm_qk_kernel(
    const float* __restrict__ nqw, const float* __restrict__ nkw,
    const float* __restrict__ Wq,  const float* __restrict__ Wk,
    float* __restrict__ qh, float* __restrict__ kh)
{
    const int wid  = (blockIdx.x * 256 + threadIdx.x) >> 5;   // 2048 waves
    const int lane = threadIdx.x & 31;
    const int b     = wid >> 7;
    const int r     = wid & 127;
    const int which = r >> 6;          // 0 = q, 1 = k
    const int r2    = r & 63;
    const int mt    = r2 >> 4;         // 0..3  (M tiles of 64)
    const int nt    = r2 & 15;         // 0..15 (N tiles of 256)
    const float* A = (which ? nkw : nqw) + (size_t)b * NSMP * CDIM;
    const float* W = which ? Wk : Wq;
    float*       D = (which ? kh : qh) + (size_t)b * NSMP * CDIM;
    wmma_tile_f32(A, CDIM, W, CDIM, D, CDIM, mt * 16, nt * 16, CDIM, lane);
}

// ---- 5a) logits = qh @ kh^T (per batch 64x64, K=256) ----
__global__ __launch_bounds__(256) void logits_kernel(
    const float* __restrict__ qh, const float* __restrict__ kh,
    float* __restrict__ logits)
{
    const int wid  = (blockIdx.x * 256 + threadIdx.x) >> 5;   // 256 waves
    const int lane = threadIdx.x & 31;
    const int b  = wid >> 4;
    const int t  = wid & 15;
    const int mt = t >> 2;
    const int nt = t & 3;
    const float* A = qh + (size_t)b * NSMP * CDIM;
    const float* B = kh + (size_t)b * NSMP * CDIM;
    float*       D = logits + (size_t)b * NSMP * NSMP;
    wmma_tile_f32(A, CDIM, B, CDIM, D, NSMP, mt * 16, nt * 16, CDIM, lane);
}

// ---- 5b) softmax (shared by both cross_attn calls) + attn@v + outputs ----
__global__ __launch_bounds__(256) void attn_kernel(
    const float* __restrict__ logits,
    const float* __restrict__ dcoor, const float* __restrict__ dif,
    const float* __restrict__ scoor, const float* __restrict__ sif,
    float* __restrict__ out2, float* __restrict__ out3)
{
    const int wid  = (blockIdx.x * 256 + threadIdx.x) >> 5;   // 1024 waves = rows
    const int lane = threadIdx.x & 31;
    const int b = wid >> 6;
    const int i = wid & 63;
    const float* lrow = logits + (size_t)b * NSMP * NSMP + (size_t)i * NSMP;

    float l0 = lrow[lane], l1 = lrow[lane + 32];
    float m = fmaxf(l0, l1);
#pragma unroll
    for (int off = 16; off > 0; off >>= 1) m = fmaxf(m, __shfl_xor(m, off, 32));
    const float e0 = expf(l0 - m), e1 = expf(l1 - m);
    float ssum = e0 + e1;
#pragma unroll
    for (int off = 16; off > 0; off >>= 1) ssum += __shfl_xor(ssum, off, 32);
    const float p0 = e0 / ssum, p1 = e1 / ssum;

#pragma unroll
    for (int c = 0; c < 3; ++c) {
        // v[b,j,c] = diff[b, (3j+c)%64, (3j+c)/64]   (transpose+reshape permutation)
        const int k0 = lane * 3 + c, k1 = (lane + 32) * 3 + c;
        const float vc0 = dcoor[b*192 + (k0 & 63)*3 + (k0 >> 6)];
        const float vc1 = dcoor[b*192 + (k1 & 63)*3 + (k1 >> 6)];
        const float vf0 = dif  [b*192 + (k0 & 63)*3 + (k0 >> 6)];
        const float vf1 = dif  [b*192 + (k1 & 63)*3 + (k1 >> 6)];
        float ac = p0 * vc0 + p1 * vc1;
        float af = p0 * vf0 + p1 * vf1;
#pragma unroll
        for (int off = 16; off > 0; off >>= 1) {
            ac += __shfl_xor(ac, off, 32);
            af += __shfl_xor(af, off, 32);
        }
        if (lane == 0) {
            out2[(b*64 + i)*3 + c] = scoor[(b*64 + i)*3 + c] + ac;
            out3[(b*64 + i)*3 + c] = sif  [(b*64 + i)*3 + c] + af;
        }
    }
}

// ======================= host launcher =======================
extern "C" void kernel_launch(void* const* d_in, const int* in_sizes, int n_in,
                              void* d_out, int out_size, void* d_ws, size_t ws_size,
                              hipStream_t stream)
{
    const float* iff  = (const float*)d_in[0];  // input_feature (B,N,3)
    const float* x    = (const float*)d_in[1];  // x (B,N,256)
    const float* coor = (const float*)d_in[2];  // coor (B,N,3)
    const float* Wq   = (const float*)d_in[3];
    const float* Wk   = (const float*)d_in[4];
    const float* gq   = (const float*)d_in[5];
    const float* bq   = (const float*)d_in[6];
    const float* gk   = (const float*)d_in[7];
    const float* bk   = (const float*)d_in[8];

    float* out1 = (float*)d_out;                         // (B,64,256)
    float* out2 = out1 + (size_t)BATCH * NSMP * CDIM;    // (B,64,3)
    float* out3 = out2 + (size_t)BATCH * NSMP * 3;       // (B,64,3)

    char* ws = (char*)d_ws;
    int*   idx    = (int*)  (ws + OFF_IDX);
    int*   nidx   = (int*)  (ws + OFF_NIDX);
    float* scoor  = (float*)(ws + OFF_SCOOR);
    float* sif    = (float*)(ws + OFF_SIF);
    float* dcoor  = (float*)(ws + OFF_DCOOR);
    float* dif    = (float*)(ws + OFF_DIF);
    float* nqw    = (float*)(ws + OFF_NQ);
    float* nkw    = (float*)(ws + OFF_NK);
    float* qh     = (float*)(ws + OFF_QH);
    float* kh     = (float*)(ws + OFF_KH);
    float* logits = (float*)(ws + OFF_LOG);

    fps_kernel      <<<BATCH, 1024, 0, stream>>>(coor, idx);
    ballq_kernel    <<<BATCH, NSMP, 0, stream>>>(coor, idx, nidx);
    gather_ln_kernel<<<BATCH * NSMP, 256, 0, stream>>>(x, coor, iff, idx, nidx,
                                                       gq, bq, gk, bk,
                                                       out1, nqw, nkw,
                                                       scoor, sif, dcoor, dif);
    gemm_qk_kernel  <<<256, 256, 0, stream>>>(nqw, nkw, Wq, Wk, qh, kh);
    logits_kernel   <<<32, 256, 0, stream>>>(qh, kh, logits);
    attn_kernel     <<<128, 256, 0, stream>>>(logits, dcoor, dif, scoor, sif, out2, out3);
}